// Canny_Net_44126493999448
// MI455X (gfx1250) — compile-verified
//
#include <hip/hip_runtime.h>
#include <hip/hip_bf16.h>

// ---------------------------------------------------------------------------
// Canny edge detector, one image (32x32) per workgroup, fully LDS-resident.
// Gaussian smoothing done as 32x32 banded-matrix matmuls on the CDNA5 tensor
// path: V_WMMA_F32_16X16X4_F32 (full f32 precision).
// ---------------------------------------------------------------------------

typedef float v2f __attribute__((ext_vector_type(2)));
typedef float v8f __attribute__((ext_vector_type(8)));

// 9-tap Gaussian (sigma=1, lw=4), normalized in f64 then rounded to f32
// (matches scipy/_gaussian_kernel1d; symmetric so reversal is a no-op).
__constant__ float c_GK[9] = {
    1.3383062e-04f, 4.4318616e-03f, 5.3991127e-02f, 2.4197145e-01f,
    3.9894347e-01f, 2.4197145e-01f, 5.3991127e-02f, 4.4318616e-03f,
    1.3383062e-04f};

// ---------------------------------------------------------------------------
// 32x32 f32 matmul in LDS using 4 waves of WMMA 16x16x4.
//   BT=false: C = A * B      BT=true: C = A * B^T
// A-layout (16x4 f32): lanes 0-15 -> M=lane, VGPR{0,1}=K{0,1};
//                      lanes 16-31 -> M=lane-16, VGPR{0,1}=K{2,3}.
// B-layout (4x16 f32): VGPR0 = rows K{0 | 2} striped over lane halves, etc.
// C/D: VGPR v holds M = v + 8*(lane>=16), N = lane&15.
// ---------------------------------------------------------------------------
template <bool BT>
__device__ __forceinline__ void mm32(const float* A, const float* Bm, float* C,
                                     int tid) {
  const int lane = tid & 31;
  const int wave = tid >> 5;          // 0..3
  const int half = lane >> 4;         // 0 or 1
  const int l16  = lane & 15;
  const int m    = ((wave & 1) << 4) + l16;   // A row
  const int n    = ((wave >> 1) << 4) + l16;  // B col
  const int koff = half << 1;

  v8f acc = {};
#pragma unroll
  for (int k0 = 0; k0 < 32; k0 += 4) {
    const int k = k0 + koff;
    v2f a, b;
    a.x = A[m * 32 + k];
    a.y = A[m * 32 + k + 1];
    if (BT) {
      b.x = Bm[n * 32 + k];
      b.y = Bm[n * 32 + k + 1];
    } else {
      b.x = Bm[k * 32 + n];
      b.y = Bm[(k + 1) * 32 + n];
    }
    acc = __builtin_amdgcn_wmma_f32_16x16x4_f32(false, a, false, b, (short)0,
                                                acc, false, false);
  }
  const int mrow = ((wave & 1) << 4) + (half << 3);
#pragma unroll
  for (int v = 0; v < 8; ++v) C[(mrow + v) * 32 + n] = acc[v];
}

// zero-padded shifted read of a 32x32 LDS image
__device__ __forceinline__ float shz(const float* M, int i, int j) {
  return (i >= 0 && i < 32 && j >= 0 && j < 32) ? M[i * 32 + j] : 0.f;
}

__device__ __forceinline__ bool nms_oct(const float* mag, int i, int j,
                                        float mg, float w, int p2i, int p2j,
                                        int p1i, int p1j, int m2i, int m2j,
                                        int m1i, int m1j) {
  float ip = shz(mag, i + p2i, j + p2j) * w + shz(mag, i + p1i, j + p1j) * (1.f - w);
  float im = shz(mag, i + m2i, j + m2j) * w + shz(mag, i + m1i, j + m1j) * (1.f - w);
  return (ip <= mg) && (im <= mg);
}

// ---------------------------------------------------------------------------
// Kernel 1 (1 block, 1024 threads): batch-independent precompute.
//   bleed = gauss(mask), er = erode3x3(mask!=0) & (mag2 of IMAGE 0 > 0)
// ---------------------------------------------------------------------------
__global__ void canny_prep(const float* __restrict__ x,
                           const float* __restrict__ mask,
                           float* __restrict__ bleed_ws,
                           unsigned char* __restrict__ er_ws) {
  __shared__ float sA[1024];  // mask, then gray
  __shared__ float sB[1024];  // temp
  __shared__ float sC[1024];  // sm
  const int tid = threadIdx.x;
  const int i = tid >> 5, j = tid & 31;

  sA[tid] = mask[tid];
  __syncthreads();

  // bleed = separable zero-padded gauss of mask
  float acc = 0.f;
#pragma unroll
  for (int t = 0; t < 9; ++t) {
    int c = j + t - 4;
    if (c >= 0 && c < 32) acc += sA[i * 32 + c] * c_GK[t];
  }
  sB[tid] = acc;
  __syncthreads();
  acc = 0.f;
#pragma unroll
  for (int t = 0; t < 9; ++t) {
    int r = i + t - 4;
    if (r >= 0 && r < 32) acc += sB[r * 32 + j] * c_GK[t];
  }
  const float bleed = acc;
  bleed_ws[tid] = bleed;

  // 3x3 erosion of mask!=0, zero border
  bool er2d = true;
#pragma unroll
  for (int di = -1; di <= 1; ++di)
#pragma unroll
    for (int dj = -1; dj <= 1; ++dj) {
      int ii = i + di, jj = j + dj;
      bool v = (ii >= 0 && ii < 32 && jj >= 0 && jj < 32) ? (sA[ii * 32 + jj] != 0.f) : false;
      er2d = er2d && v;
    }

  // image 0 gray
  float r0 = x[tid] * 0.5f + 0.5f;
  float g0 = x[1024 + tid] * 0.5f + 0.5f;
  float b0 = x[2048 + tid] * 0.5f + 0.5f;
  __syncthreads();  // done reading sA/sB
  sA[tid] = r0 * 0.299f + g0 * 0.587f + b0 * 0.114f;
  __syncthreads();

  // gauss(gray) / (bleed + 1e-12)
  acc = 0.f;
#pragma unroll
  for (int t = 0; t < 9; ++t) {
    int c = j + t - 4;
    if (c >= 0 && c < 32) acc += sA[i * 32 + c] * c_GK[t];
  }
  sB[tid] = acc;
  __syncthreads();
  acc = 0.f;
#pragma unroll
  for (int t = 0; t < 9; ++t) {
    int r = i + t - 4;
    if (r >= 0 && r < 32) acc += sB[r * 32 + j] * c_GK[t];
  }
  sC[tid] = acc / (bleed + 1e-12f);
  __syncthreads();

  // Sobel (edge padding): W-derivative then H-smooth -> jsob
  sB[tid] = sC[i * 32 + (j < 31 ? j + 1 : 31)] - sC[i * 32 + (j > 0 ? j - 1 : 0)];
  __syncthreads();
  float jsob = sB[(i > 0 ? i - 1 : 0) * 32 + j] + 2.f * sB[i * 32 + j] +
               sB[(i < 31 ? i + 1 : 31) * 32 + j];
  __syncthreads();
  sB[tid] = sC[(i < 31 ? i + 1 : 31) * 32 + j] - sC[(i > 0 ? i - 1 : 0) * 32 + j];
  __syncthreads();
  float isob = sB[i * 32 + (j > 0 ? j - 1 : 0)] + 2.f * sB[i * 32 + j] +
               sB[i * 32 + (j < 31 ? j + 1 : 31)];
  float mag2 = isob * isob + jsob * jsob;

  er_ws[tid] = (er2d && (mag2 > 0.f)) ? 1 : 0;
}

// ---------------------------------------------------------------------------
// Kernel 2: one image per block, 128 threads (4 waves), 8 px/thread.
// ---------------------------------------------------------------------------
__global__ void __launch_bounds__(128) canny_main(
    const float* __restrict__ x, const float* __restrict__ bleed_ws,
    const unsigned char* __restrict__ er_ws, float* __restrict__ out) {
  __shared__ float sG[1024];    // banded Gaussian matrix
  __shared__ float sGray[1024];
  __shared__ float sT1[1024];   // gray * G^T
  __shared__ float sSm[1024];   // smoothed / bleed
  __shared__ float sTmp[1024];
  __shared__ float sI[1024], sJ[1024], sMag[1024];
  __shared__ unsigned char sLow[1024], sCur[1024];
  __shared__ int sChanged;

  const int tid = threadIdx.x;
  const size_t img = blockIdx.x;
  const float* xr = x + img * 3072;

  // G[j][c] = gk[c-j+4] (|c-j|<=4), and luma of (x*0.5+0.5)
#pragma unroll
  for (int q = 0; q < 8; ++q) {
    int p = tid + q * 128;
    int jj = p >> 5, cc = p & 31;
    int d = cc - jj;
    sG[p] = (d >= -4 && d <= 4) ? c_GK[d + 4] : 0.f;
    float r = xr[p] * 0.5f + 0.5f;
    float g = xr[p + 1024] * 0.5f + 0.5f;
    float b = xr[p + 2048] * 0.5f + 0.5f;
    sGray[p] = r * 0.299f + g * 0.587f + b * 0.114f;
  }
  __syncthreads();

  // Separable zero-padded Gaussian as two WMMA matmuls:
  mm32<true>(sGray, sG, sT1, tid);   // T1 = Gray * G^T   (W-conv)
  __syncthreads();
  mm32<false>(sG, sT1, sSm, tid);    // smRaw = G * T1    (H-conv)
  __syncthreads();

#pragma unroll
  for (int q = 0; q < 8; ++q) {
    int p = tid + q * 128;
    sSm[p] = sSm[p] / (bleed_ws[p] + 1e-12f);
  }
  __syncthreads();

  // Sobel, edge (replicate) padding.  jsob = Hsmooth(Wderiv(sm))
#pragma unroll
  for (int q = 0; q < 8; ++q) {
    int p = tid + q * 128, i = p >> 5, j = p & 31;
    sTmp[p] = sSm[i * 32 + (j < 31 ? j + 1 : 31)] - sSm[i * 32 + (j > 0 ? j - 1 : 0)];
  }
  __syncthreads();
#pragma unroll
  for (int q = 0; q < 8; ++q) {
    int p = tid + q * 128, i = p >> 5, j = p & 31;
    sJ[p] = sTmp[(i > 0 ? i - 1 : 0) * 32 + j] + 2.f * sTmp[p] +
            sTmp[(i < 31 ? i + 1 : 31) * 32 + j];
  }
  __syncthreads();
  // isob = Wsmooth(Hderiv(sm))
#pragma unroll
  for (int q = 0; q < 8; ++q) {
    int p = tid + q * 128, i = p >> 5, j = p & 31;
    sTmp[p] = sSm[(i < 31 ? i + 1 : 31) * 32 + j] - sSm[(i > 0 ? i - 1 : 0) * 32 + j];
  }
  __syncthreads();
#pragma unroll
  for (int q = 0; q < 8; ++q) {
    int p = tid + q * 128, i = p >> 5, j = p & 31;
    sI[p] = sTmp[i * 32 + (j > 0 ? j - 1 : 0)] + 2.f * sTmp[p] +
            sTmp[i * 32 + (j < 31 ? j + 1 : 31)];
    sMag[p] = sqrtf(sI[p] * sI[p] + sJ[p] * sJ[p] + 1e-9f);
  }
  __syncthreads();

  // Non-max suppression (octant order matches reference overwrite semantics)
#pragma unroll
  for (int q = 0; q < 8; ++q) {
    int p = tid + q * 128, i = p >> 5, j = p & 31;
    float is = sI[p], js = sJ[p], mg = sMag[p];
    float ai = fabsf(is), aj = fabsf(js);
    bool same = ((is >= 0.f) && (js >= 0.f)) || ((is <= 0.f) && (js <= 0.f));
    bool opp  = ((is <= 0.f) && (js >= 0.f)) || ((is >= 0.f) && (js <= 0.f));
    float ai_s = ai > 0.f ? ai : 1.f;
    float aj_s = aj > 0.f ? aj : 1.f;
    bool lm = false;
    if (er_ws[p]) {
      if (same && ai >= aj)
        lm = nms_oct(sMag, i, j, mg, aj / (ai + 1e-9f), 1, 1, 1, 0, -1, -1, -1, 0);
      if (same && ai <= aj)
        lm = nms_oct(sMag, i, j, mg, ai / aj_s, 1, 1, 0, 1, -1, -1, 0, -1);
      if (opp && ai <= aj)
        lm = nms_oct(sMag, i, j, mg, ai / aj_s, -1, 1, 0, 1, 1, -1, 0, -1);
      if (opp && ai >= aj)
        lm = nms_oct(sMag, i, j, mg, aj / ai_s, -1, 1, -1, 0, 1, -1, 1, 0);
    }
    sLow[p] = (lm && mg >= 0.1f) ? 1 : 0;
    sCur[p] = (lm && mg >= 0.2f) ? 1 : 0;  // high seed
  }
  if (tid == 0) sChanged = 0;
  __syncthreads();

  // Hysteresis: fixpoint of cur = dilate8(cur) & low (monotone, <=1024 steps)
  for (int it = 0; it < 1025; ++it) {
    unsigned char nv[8];
#pragma unroll
    for (int q = 0; q < 8; ++q) {
      int p = tid + q * 128, i = p >> 5, j = p & 31;
      bool any = false;
#pragma unroll
      for (int di = -1; di <= 1; ++di)
#pragma unroll
        for (int dj = -1; dj <= 1; ++dj) {
          int ii = i + di, jj = j + dj;
          if (ii >= 0 && ii < 32 && jj >= 0 && jj < 32)
            any = any || (sCur[ii * 32 + jj] != 0);
        }
      nv[q] = (any && sLow[p]) ? 1 : 0;
    }
    __syncthreads();  // all reads of sCur done; sChanged==0 visible
    bool ch = false;
#pragma unroll
    for (int q = 0; q < 8; ++q) {
      int p = tid + q * 128;
      if (nv[q] != sCur[p]) { sCur[p] = nv[q]; ch = true; }
    }
    if (ch) sChanged = 1;
    __syncthreads();  // writes done
    int chg = sChanged;
    __syncthreads();  // everyone has read chg; safe for tid0 to reset
    if (tid == 0) sChanged = 0;
    if (!chg) break;  // uniform
  }

  // out = 2*final - 1
#pragma unroll
  for (int q = 0; q < 8; ++q) {
    int p = tid + q * 128;
    out[img * 1024 + p] = sCur[p] ? 1.f : -1.f;
  }
}

extern "C" void kernel_launch(void* const* d_in, const int* in_sizes, int n_in,
                              void* d_out, int out_size, void* d_ws,
                              size_t ws_size, hipStream_t stream) {
  const float* x = (const float*)d_in[0];       // (B,3,32,32) f32
  const float* mask = (const float*)d_in[1];    // (1,1,32,32) f32
  float* bleed_ws = (float*)d_ws;               // 1024 f32
  unsigned char* er_ws = (unsigned char*)(bleed_ws + 1024);  // 1024 u8
  const int B = in_sizes[0] / 3072;             // 8192

  canny_prep<<<1, 1024, 0, stream>>>(x, mask, bleed_ws, er_ws);
  canny_main<<<B, 128, 0, stream>>>(x, bleed_ws, er_ws, (float*)d_out);
}